// TransformerAttentionModule_37907381354768
// MI455X (gfx1250) — compile-verified
//
#include <hip/hip_runtime.h>
#include <math.h>

#define NNODES   50000
#define DIMX     128
#define NHEADS   8
#define HDIM     16
#define NEDGES   1600000
#define TILES_M  (NNODES / 16)          // 3125, exact
#define ND       (NNODES * DIMX)        // 6,400,000 floats
#define NH       (NNODES * NHEADS)      // 400,000 floats
#define EH       (NEDGES * NHEADS)      // 12,800,000 threads

#define AROW     132                    // padded LDS row stride (floats): bank-conflict-free
#define BOFF     (16 * AROW)            // offset of W tile region in LDS
#define LDS_FLOATS (BOFF + 128 * AROW)  // 2112 + 16896 = 19008 floats = 76032 B

typedef float v2f __attribute__((ext_vector_type(2)));
typedef float v8f __attribute__((ext_vector_type(8)));
typedef int   v4i __attribute__((ext_vector_type(4)));

#define AS1 __attribute__((address_space(1)))
#define AS3 __attribute__((address_space(3)))

#if defined(__AMDGCN__) && __has_builtin(__builtin_amdgcn_global_load_async_to_lds_b128)
#define HAVE_ASYNC_LDS 1
#else
#define HAVE_ASYNC_LDS 0
#endif

// 16-byte global -> LDS copy (async data-mover path when available)
__device__ __forceinline__ void cp16_g2l(const float* g, float* l)
{
#if HAVE_ASYNC_LDS
    __builtin_amdgcn_global_load_async_to_lds_b128((AS1 v4i*)g, (AS3 v4i*)l,
                                                   /*offset=*/0, /*cpol=*/0);
#else
    *(float4*)l = *(const float4*)g;
#endif
}

__device__ __forceinline__ void wait_g2l()
{
#if HAVE_ASYNC_LDS
#if __has_builtin(__builtin_amdgcn_s_wait_asynccnt)
    __builtin_amdgcn_s_wait_asynccnt(0);
#else
    asm volatile("s_wait_asynccnt 0" ::: "memory");
#endif
#endif
}

// ---------------------------------------------------------------------------
// Workspace init: AGG = 0, Z = 0, M = -inf   (ws is poisoned before timing)
// ---------------------------------------------------------------------------
__global__ __launch_bounds__(256)
void gat_init(float* __restrict__ agg, float* __restrict__ mx, float* __restrict__ z)
{
    const int gid = blockIdx.x * blockDim.x + threadIdx.x;
    if (gid < ND) agg[gid] = 0.0f;
    if (gid < NH) {
        mx[gid] = -INFINITY;
        z[gid]  = 0.0f;
    }
}

// ---------------------------------------------------------------------------
// Y = X @ W^T + bias   (X: [rows,128], W: [128,128] row-major, Y: [rows,128])
// Block = 8 waves; stages A tile (16x128) + full W (128x128) into padded LDS
// (row stride 132 floats -> ds_load_b64 is bank-conflict-free), then each
// wave computes one 16x16 tile of Y via V_WMMA_F32_16X16X4_F32 fed from LDS.
// A (16x4 f32): lane r=lane%16 -> row m0+r; VGPR pair holds K = 2*(lane/16)+{0,1}
// B (4x16 f32): lane r -> col n0+r (B[k][n] = W[n][k] since Y = X W^T)
// C/D (16x16 f32): VGPR v, lanes 0-15 -> M=v, lanes 16-31 -> M=v+8; N = lane%16
// ---------------------------------------------------------------------------
__global__ __launch_bounds__(256)
void gat_gemm_xWT(const float* __restrict__ X, const float* __restrict__ W,
                  const float* __restrict__ bias, float* __restrict__ Y)
{
    __shared__ float lds[LDS_FLOATS];

    const int lane = threadIdx.x & 31;
    const int wave = threadIdx.x >> 5;      // 8 waves -> 8 column tiles (128 cols)
    const int m0   = blockIdx.x << 4;
    const int n0   = wave << 4;
    const int r    = lane & 15;
    const int kh   = lane >> 4;             // 0 or 1: which K-pair of the quad

    // ---- stage A tile: 16 rows x 128 floats = 512 float4s over 256 threads
#pragma unroll
    for (int j = threadIdx.x; j < 512; j += 256) {
        const int row = j >> 5;             // 32 float4 per row
        const int c4  = j & 31;
        cp16_g2l(X + (size_t)(m0 + row) * DIMX + c4 * 4,
                 &lds[row * AROW + c4 * 4]);
    }
    // ---- stage W: 128 rows x 128 floats = 4096 float4s over 256 threads
#pragma unroll
    for (int j = threadIdx.x; j < 4096; j += 256) {
        const int row = j >> 5;
        const int c4  = j & 31;
        cp16_g2l(W + (size_t)row * DIMX + c4 * 4,
                 &lds[BOFF + row * AROW + c4 * 4]);
    }
    wait_g2l();
    __syncthreads();

    const float* __restrict__ la = &lds[r * AROW + 2 * kh];
    const float* __restrict__ lb = &lds[BOFF + (n0 + r) * AROW + 2 * kh];

    v8f acc = {};
#pragma unroll
    for (int k0 = 0; k0 < DIMX; k0 += 4) {
        v2f a = *(const v2f*)(la + k0);
        v2f b = *(const v2f*)(lb + k0);
        acc = __builtin_amdgcn_wmma_f32_16x16x4_f32(
            /*neg_a=*/false, a, /*neg_b=*/false, b,
            /*c_mod=*/(short)0, acc, /*reuse_a=*/false, /*reuse_b=*/false);
    }

    const float bn = bias[n0 + r];
#pragma unroll
    for (int v = 0; v < 8; ++v) {
        const int m = m0 + v + (kh << 3);
        Y[(size_t)m * DIMX + (n0 + r)] = acc[v] + bn;
    }
}

// ---------------------------------------------------------------------------
// Float atomic max via monotone int orderings (initial value must be -inf).
// Non-negative floats order like signed ints; negative like reversed unsigned.
// ---------------------------------------------------------------------------
__device__ __forceinline__ void atomicMaxF32(float* addr, float val)
{
    if (val >= 0.0f) {
        atomicMax((int*)addr, __float_as_int(val));
    } else {
        atomicMin((unsigned int*)addr, __float_as_uint(val));
    }
}

// ---------------------------------------------------------------------------
// scores[e,h] = (q[src[e],h,:] . k[dst[e],h,:]) / sqrt(16); running max per (dst,h)
// ---------------------------------------------------------------------------
__global__ __launch_bounds__(256)
void gat_scores(const float* __restrict__ Q, const float* __restrict__ K,
                const int* __restrict__ src, const int* __restrict__ dst,
                float* __restrict__ S, float* __restrict__ Mx)
{
    const int gid = blockIdx.x * blockDim.x + threadIdx.x;
    if (gid >= EH) return;
    const int e = gid >> 3;
    const int h = gid & 7;
    const int s = src[e];
    const int d = dst[e];

    const float4* __restrict__ qp = (const float4*)(Q + (size_t)s * DIMX + h * HDIM);
    const float4* __restrict__ kp = (const float4*)(K + (size_t)d * DIMX + h * HDIM);

    float acc = 0.0f;
#pragma unroll
    for (int i = 0; i < 4; ++i) {
        const float4 qv = qp[i];
        const float4 kv = kp[i];
        acc += qv.x * kv.x + qv.y * kv.y + qv.z * kv.z + qv.w * kv.w;
    }
    acc *= 0.25f;                       // 1/sqrt(HEAD_DIM)

    S[gid] = acc;
    atomicMaxF32(&Mx[d * NHEADS + h], acc);
}

// ---------------------------------------------------------------------------
// e[e,h] = exp(s - m[dst,h]);  z[dst,h] += e   (e stored back into S)
// ---------------------------------------------------------------------------
__global__ __launch_bounds__(256)
void gat_exp(const int* __restrict__ dst, const float* __restrict__ Mx,
             float* __restrict__ S, float* __restrict__ Z)
{
    const int gid = blockIdx.x * blockDim.x + threadIdx.x;
    if (gid >= EH) return;
    const int e = gid >> 3;
    const int h = gid & 7;
    const int d = dst[e];

    const float ev = __expf(S[gid] - Mx[d * NHEADS + h]);
    S[gid] = ev;
    atomicAdd(&Z[d * NHEADS + h], ev);
}

// ---------------------------------------------------------------------------
// agg[dst,h,:] += v[src,h,:] * (e / z[dst,h])   (L2-resident scatter-add)
// ---------------------------------------------------------------------------
__global__ __launch_bounds__(256)
void gat_agg(const float* __restrict__ V, const int* __restrict__ src,
             const int* __restrict__ dst, const float* __restrict__ S,
             const float* __restrict__ Z, float* __restrict__ AGG)
{
    const int gid = blockIdx.x * blockDim.x + threadIdx.x;
    if (gid >= EH) return;
    const int e = gid >> 3;
    const int h = gid & 7;
    const int s = src[e];
    const int d = dst[e];

    const float p = S[gid] / Z[d * NHEADS + h];

    const float4* __restrict__ vp = (const float4*)(V + (size_t)s * DIMX + h * HDIM);
    float* __restrict__ ap = AGG + (size_t)d * DIMX + h * HDIM;

#pragma unroll
    for (int i = 0; i < 4; ++i) {
        const float4 vv = vp[i];
        atomicAdd(&ap[4 * i + 0], vv.x * p);
        atomicAdd(&ap[4 * i + 1], vv.y * p);
        atomicAdd(&ap[4 * i + 2], vv.z * p);
        atomicAdd(&ap[4 * i + 3], vv.w * p);
    }
}

// ---------------------------------------------------------------------------
extern "C" void kernel_launch(void* const* d_in, const int* in_sizes, int n_in,
                              void* d_out, int out_size, void* d_ws, size_t ws_size,
                              hipStream_t stream)
{
    const float* x   = (const float*)d_in[0];
    const int*   src = (const int*)  d_in[1];
    const int*   dst = (const int*)  d_in[2];
    const float* Wq  = (const float*)d_in[3];
    const float* bq  = (const float*)d_in[4];
    const float* Wk  = (const float*)d_in[5];
    const float* bk  = (const float*)d_in[6];
    const float* Wv  = (const float*)d_in[7];
    const float* bv  = (const float*)d_in[8];
    const float* Wo  = (const float*)d_in[9];
    const float* bo  = (const float*)d_in[10];
    float* out = (float*)d_out;

    // Workspace layout (floats): Q | K | V | AGG | M | Z | S  ~= 157 MB
    float* Q   = (float*)d_ws;
    float* K   = Q   + ND;
    float* V   = K   + ND;
    float* AGG = V   + ND;
    float* Mx  = AGG + ND;
    float* Z   = Mx  + NH;
    float* S   = Z   + NH;

    // 1) init accumulators
    gat_init<<<(ND + 255) / 256, 256, 0, stream>>>(AGG, Mx, Z);

    // 2) Q/K/V projections (WMMA fp32, LDS-staged operands)
    gat_gemm_xWT<<<TILES_M, 256, 0, stream>>>(x, Wq, bq, Q);
    gat_gemm_xWT<<<TILES_M, 256, 0, stream>>>(x, Wk, bk, K);
    gat_gemm_xWT<<<TILES_M, 256, 0, stream>>>(x, Wv, bv, V);

    // 3) edge scores + segment max
    const int eh_blocks = (EH + 255) / 256;   // 50000, exact
    gat_scores<<<eh_blocks, 256, 0, stream>>>(Q, K, src, dst, S, Mx);

    // 4) exp + segment sum
    gat_exp<<<eh_blocks, 256, 0, stream>>>(dst, Mx, S, Z);

    // 5) weighted scatter-add of messages
    gat_agg<<<eh_blocks, 256, 0, stream>>>(V, src, dst, S, Z, AGG);

    // 6) output projection (WMMA fp32, LDS-staged operands)
    gat_gemm_xWT<<<TILES_M, 256, 0, stream>>>(AGG, Wo, bo, out);
}